// MMD_29729763623098
// MI455X (gfx1250) — compile-verified
//
#include <hip/hip_runtime.h>
#include <math.h>

// ---------------------------------------------------------------------------
// Problem constants (from reference): B=2, N_WAY=5, K_SHOT=5, C=64, H=W=10,
// Q=75, PROJ=64, NHEAD=8, dh=8. n=100 positions per image.
// ---------------------------------------------------------------------------
#define NIMG      200     // 2*25 support images + 2*75 query images
#define NSUP_IMG  50
#define C_DIM     64
#define NPOS      100
#define NS_PAD    512     // 500 support rows padded to 32 tiles of 16
#define M_PAD     112     // 100 query rows padded to 7 tiles of 16

// Workspace layout (float offsets)
#define KQV_OFF   0         // 3*200*64*100 = 3,840,000  (K,Q,V projections)
#define ATT_OFF   3840000   // 200*64*100   = 1,280,000  (attention output, pre-residual)
#define SUPN_OFF  5120000   // 10*512*64    =   327,680  (normalized support, padded)
#define QRYN_OFF  5447680   // 150*112*64   = 1,075,200  (normalized query, padded)
#define SNS_OFF   6522880   // 10*512       =     5,120  (support row ||.||^2)
#define QN_OFF    6528000   // 150*112      =    16,800  (query row ||.||^2)
#define ACCSS_OFF 6544800   // 10
#define ACCQQ_OFF 6544810   // 150
#define ACCQS_OFF 6544960   // 750

typedef float v2f __attribute__((ext_vector_type(2)));
typedef float v8f __attribute__((ext_vector_type(8)));

__device__ __forceinline__ const float* img_x(const float* sup, const float* qry, int img) {
  return (img < NSUP_IMG) ? (sup + img * C_DIM * NPOS)
                          : (qry + (img - NSUP_IMG) * C_DIM * NPOS);
}

// ---------------------------------------------------------------------------
// Kernel 0: init padded buffers / accumulators.
// Padded rows stay zero; padded squared-norms = 1e30 so exp(-alpha*d) == 0.
// ---------------------------------------------------------------------------
__global__ void init_kernel(float* __restrict__ ws) {
  int i = blockIdx.x * blockDim.x + threadIdx.x;
  int stride = gridDim.x * blockDim.x;
  for (int k = i; k < 327680 + 1075200; k += stride) ws[SUPN_OFF + k] = 0.f;
  for (int k = i; k < 5120 + 16800;     k += stride) ws[SNS_OFF  + k] = 1e30f;
  for (int k = i; k < 910;              k += stride) ws[ACCSS_OFF + k] = 0.f;
}

// ---------------------------------------------------------------------------
// Kernel 1: K/Q/V projections via WMMA f32 16x16x4.
// 200 images x 3 mats x (4x7) tiles = 16800 tile-tasks, one per wave.
// Branchless padding: clamp column, multiply by 0/1 mask (no EXEC juggling
// inside the unrolled K loop).
// ---------------------------------------------------------------------------
__global__ void proj_wmma_kernel(const float* __restrict__ sup, const float* __restrict__ qry,
                                 const float* __restrict__ Wk, const float* __restrict__ Wq,
                                 const float* __restrict__ Wv, float* __restrict__ kqv) {
  int wave = (blockIdx.x * blockDim.x + threadIdx.x) >> 5;   // wave-uniform
  int lane = threadIdx.x & 31;
  if (wave >= NIMG * 3 * 28) return;
  int img  = wave / 84;
  int r    = wave % 84;
  int mat  = r / 28;
  int tile = r % 28;
  int ti = tile / 7, tj = tile % 7;
  const float* W  = (mat == 0) ? Wk : (mat == 1) ? Wq : Wv;
  const float* xp = img_x(sup, qry, img);

  int rowA = ti * 16 + (lane & 15);          // W output-channel row
  int col  = tj * 16 + (lane & 15);          // spatial position (may be padded)
  int kh   = (lane >> 4) * 2;                // K sub-offset for this half-wave
  int colc = (col < NPOS) ? col : (NPOS - 1);
  float msk = (col < NPOS) ? 1.f : 0.f;

  v8f acc = {0.f, 0.f, 0.f, 0.f, 0.f, 0.f, 0.f, 0.f};
#pragma unroll
  for (int s = 0; s < 16; ++s) {
    int cs = s * 4 + kh;
    v2f a = *(const v2f*)(W + rowA * 64 + cs);     // A[m][k], A[m][k+1]
    v2f b;                                         // B[k][n], B[k+1][n]
    b.x = xp[cs * NPOS + colc] * msk;
    b.y = xp[(cs + 1) * NPOS + colc] * msk;
    acc = __builtin_amdgcn_wmma_f32_16x16x4_f32(false, a, false, b, (short)0, acc,
                                                false, false);
  }
  if (col < NPOS) {
    float* outp = kqv + (mat * NIMG + img) * C_DIM * NPOS;
    int m0 = ti * 16 + (lane >> 4) * 8;            // D: VGPR r -> M = r + 8*(lane/16)
#pragma unroll
    for (int rr = 0; rr < 8; ++rr)
      outp[(m0 + rr) * NPOS + col] = acc[rr];
  }
}

// ---------------------------------------------------------------------------
// Kernel 2: per-(image, head) softmax attention (dh=8, n=100), flash style.
// ---------------------------------------------------------------------------
__global__ void attn_kernel(const float* __restrict__ kqv, float* __restrict__ att) {
  int img = blockIdx.x >> 3;
  int h   = blockIdx.x & 7;
  __shared__ float kh_s[8][NPOS], qh_s[8][NPOS], vh_s[8][NPOS];
  const float* kb = kqv + (0 * NIMG + img) * C_DIM * NPOS + h * 8 * NPOS;
  const float* qb = kqv + (1 * NIMG + img) * C_DIM * NPOS + h * 8 * NPOS;
  const float* vb = kqv + (2 * NIMG + img) * C_DIM * NPOS + h * 8 * NPOS;
  for (int t = threadIdx.x; t < 8 * NPOS; t += blockDim.x) {
    int d = t / NPOS, n = t % NPOS;
    kh_s[d][n] = kb[t];
    qh_s[d][n] = qb[t];
    vh_s[d][n] = vb[t];
  }
  __syncthreads();
  int i = threadIdx.x;
  if (i < NPOS) {
    const float sc = 0.35355339059327373f;   // 1/sqrt(dh=8)
    float qv[8];
#pragma unroll
    for (int d = 0; d < 8; ++d) qv[d] = qh_s[d][i] * sc;
    float m = -3.4e38f, l = 0.f;
    float acc[8] = {0.f, 0.f, 0.f, 0.f, 0.f, 0.f, 0.f, 0.f};
    for (int j = 0; j < NPOS; ++j) {
      float s = 0.f;
#pragma unroll
      for (int d = 0; d < 8; ++d) s += qv[d] * kh_s[d][j];
      float mn = fmaxf(m, s);
      float w0 = __expf(m - mn);
      float p  = __expf(s - mn);
      l = l * w0 + p;
#pragma unroll
      for (int d = 0; d < 8; ++d) acc[d] = acc[d] * w0 + p * vh_s[d][j];
      m = mn;
    }
    float inv = 1.f / l;
    float* ap = att + img * C_DIM * NPOS;
#pragma unroll
    for (int d = 0; d < 8; ++d) ap[(h * 8 + d) * NPOS + i] = acc[d] * inv;
  }
}

// ---------------------------------------------------------------------------
// Kernel 3: residual + channel-centering + L2 norm; scatter to padded
// channels-last layouts; store per-row squared norm.
// ---------------------------------------------------------------------------
__global__ void norm_kernel(const float* __restrict__ sup, const float* __restrict__ qry,
                            float* __restrict__ ws) {
  int gid = blockIdx.x * blockDim.x + threadIdx.x;
  if (gid >= NIMG * NPOS) return;
  int img = gid / NPOS, pos = gid % NPOS;
  const float* xp = img_x(sup, qry, img);
  const float* ap = ws + ATT_OFF + img * C_DIM * NPOS;
  float y[64];
  float mean = 0.f;
#pragma unroll
  for (int c = 0; c < 64; ++c) { y[c] = ap[c * NPOS + pos] + xp[c * NPOS + pos]; mean += y[c]; }
  mean *= (1.f / 64.f);
  float ss = 0.f;
#pragma unroll
  for (int c = 0; c < 64; ++c) { y[c] -= mean; ss += y[c] * y[c]; }
  float inv = rsqrtf(ss + 1e-12f);
  float* dst; float* snp;
  if (img < NSUP_IMG) {
    int b = img / 25, s5 = img % 25, way = s5 / 5, shot = s5 % 5;
    int pr = b * 5 + way;
    int row = shot * NPOS + pos;
    dst = ws + SUPN_OFF + (pr * NS_PAD + row) * 64;
    snp = ws + SNS_OFF + pr * NS_PAD + row;
  } else {
    int qimg = img - NSUP_IMG;
    int pr = (qimg / 75) * 75 + (qimg % 75);
    dst = ws + QRYN_OFF + (pr * M_PAD + pos) * 64;
    snp = ws + QN_OFF + pr * M_PAD + pos;
  }
#pragma unroll
  for (int c = 0; c < 64; ++c) dst[c] = y[c] * inv;
  *snp = ss * inv * inv;
}

// ---------------------------------------------------------------------------
// Kernel 4: WMMA Gram strips fused with multi-scale Gaussian kernel sums.
// One wave computes a 16 x (16*NT) strip: A tile loaded once per K-step and
// reused across NT WMMAs (NT=4 for the ss/qs shapes, tilesN % 4 == 0).
// sum_a exp(-a*d) = e + e^2 + e^4 + e^8 + e^16, e = exp(-d/2).
// ---------------------------------------------------------------------------
template <int NT>
__device__ __forceinline__ void gram_strip(const float* __restrict__ A,
                                           const float* __restrict__ B,
                                           const float* __restrict__ snA,
                                           const float* __restrict__ snB,
                                           int ti, int tj0, int lane, int excl,
                                           float* __restrict__ accp) {
  const float* ar = A + (ti * 16 + (lane & 15)) * 64;
  int kh = (lane >> 4) * 2;
  int i0 = ti * 16 + (lane >> 4) * 8;
  const float* br[NT];
#pragma unroll
  for (int u = 0; u < NT; ++u) br[u] = B + ((tj0 + u) * 16 + (lane & 15)) * 64;

  // Prefetch the epilogue norms while the WMMA loop runs (global_prefetch_b8).
  __builtin_prefetch(snA + i0, 0, 0);
  __builtin_prefetch(snB + tj0 * 16 + (lane & 15), 0, 0);

  v8f g[NT];
#pragma unroll
  for (int u = 0; u < NT; ++u) g[u] = (v8f){0.f, 0.f, 0.f, 0.f, 0.f, 0.f, 0.f, 0.f};

#pragma unroll
  for (int s = 0; s < 16; ++s) {
    v2f a = *(const v2f*)(ar + s * 4 + kh);        // shared A fragment
#pragma unroll
    for (int u = 0; u < NT; ++u) {
      v2f b = *(const v2f*)(br[u] + s * 4 + kh);
      g[u] = __builtin_amdgcn_wmma_f32_16x16x4_f32(false, a, false, b, (short)0, g[u],
                                                   false, false);
    }
  }

  float sum = 0.f;
#pragma unroll
  for (int u = 0; u < NT; ++u) {
    int j = (tj0 + u) * 16 + (lane & 15);
    float snj = snB[j];
#pragma unroll
    for (int rr = 0; rr < 8; ++rr) {
      int i = i0 + rr;
      float d = fmaxf(snA[i] + snj - 2.f * g[u][rr], 0.f);
      float e = __expf(-0.5f * d);                 // padded rows: sn=1e30 -> e=0
      float e2 = e * e, e4 = e2 * e2, e8 = e4 * e4, e16 = e8 * e8;
      float kv = e + e2 + e4 + e8 + e16;
      kv = (excl && i == j) ? 0.f : kv;            // "sum minus trace"
      sum += kv;
    }
  }
  for (int off = 16; off > 0; off >>= 1) sum += __shfl_xor(sum, off, 32);
  if (lane == 0) atomicAdd(accp, sum);
}

__global__ void gram_kernel(const float* __restrict__ supN, const float* __restrict__ snS,
                            const float* __restrict__ qryN, const float* __restrict__ qn,
                            float* __restrict__ acc, int mode) {
  int lane = threadIdx.x & 31;
  int t = blockIdx.y * 8 + (threadIdx.x >> 5);   // strip id, wave-uniform
  int p = blockIdx.x;
  if (mode == 0) {
    // k_ss: 32 x 32 tiles -> 32 x 8 strips of width 4; grid.y = 32
    if (t >= 32 * 8) return;
    const float* A  = supN + p * NS_PAD * 64;
    const float* sn = snS + p * NS_PAD;
    gram_strip<4>(A, A, sn, sn, t >> 3, (t & 7) * 4, lane, 1, acc + p);
  } else if (mode == 1) {
    // k_qq: 7 x 7 tiles, single-tile strips; grid.y = 7
    if (t >= 49) return;
    const float* A  = qryN + p * M_PAD * 64;
    const float* sn = qn + p * M_PAD;
    gram_strip<1>(A, A, sn, sn, t / 7, t % 7, lane, 1, acc + p);
  } else {
    // k_qs: 7 x 32 tiles -> 7 x 8 strips of width 4; grid.y = 7
    if (t >= 7 * 8) return;
    int bq = p / 5, w = p % 5, b = bq / 75;
    gram_strip<4>(qryN + bq * M_PAD * 64, supN + (b * 5 + w) * NS_PAD * 64,
                  qn + bq * M_PAD, snS + (b * 5 + w) * NS_PAD,
                  t >> 3, (t & 7) * 4, lane, 0, acc + p);
  }
}

// ---------------------------------------------------------------------------
// Kernel 5: assemble MMD terms, log-softmax over 5 ways, NLL mean.
// ---------------------------------------------------------------------------
__global__ void finalize_kernel(const float* __restrict__ ws, const int* __restrict__ qy,
                                float* __restrict__ out) {
  __shared__ float red[256];
  int tid = threadIdx.x;
  float contrib = 0.f;
  if (tid < 150) {
    const float* acc_ss = ws + ACCSS_OFF;
    const float* acc_qq = ws + ACCQQ_OFF;
    const float* acc_qs = ws + ACCQS_OFF;
    int b = tid / 75;
    float mq = acc_qq[tid] * (1.f / (100.f * 99.f));
    float lg[5];
    float mx = -3.4e38f;
#pragma unroll
    for (int w = 0; w < 5; ++w) {
      float ms  = acc_ss[b * 5 + w] * (1.f / (500.f * 499.f));
      float msq = (-2.f / (100.f * 500.f)) * acc_qs[tid * 5 + w];
      float mmd = ms + mq + msq;
      lg[w] = -mmd / 12.5f;
      mx = fmaxf(mx, lg[w]);
    }
    float se = 0.f;
#pragma unroll
    for (int w = 0; w < 5; ++w) se += __expf(lg[w] - mx);
    float lse = mx + __logf(se);
    int y = qy[tid];
    contrib = -(lg[y] - lse) * (1.f / 150.f);
  }
  red[tid] = contrib;
  __syncthreads();
  for (int s = 128; s > 0; s >>= 1) {
    if (tid < s) red[tid] += red[tid + s];
    __syncthreads();
  }
  if (tid == 0) out[0] = red[0];
}

// ---------------------------------------------------------------------------
extern "C" void kernel_launch(void* const* d_in, const int* in_sizes, int n_in,
                              void* d_out, int out_size, void* d_ws, size_t ws_size,
                              hipStream_t stream) {
  const float* sup = (const float*)d_in[0];   // (2,25,64,10,10)
  const float* qry = (const float*)d_in[1];   // (2,75,64,10,10)
  // d_in[2] = support_y (unused by the loss)
  const int*   qy  = (const int*)d_in[3];     // (2,75)
  const float* Wk  = (const float*)d_in[4];
  const float* Wq  = (const float*)d_in[5];
  const float* Wv  = (const float*)d_in[6];
  float* ws  = (float*)d_ws;                  // needs ~26.2 MB
  float* out = (float*)d_out;

  init_kernel<<<512, 256, 0, stream>>>(ws);

  // 16800 WMMA tile-tasks, 8 waves/block
  proj_wmma_kernel<<<2100, 256, 0, stream>>>(sup, qry, Wk, Wq, Wv, ws + KQV_OFF);

  attn_kernel<<<NIMG * 8, 128, 0, stream>>>(ws + KQV_OFF, ws + ATT_OFF);

  norm_kernel<<<(NIMG * NPOS + 255) / 256, 256, 0, stream>>>(sup, qry, ws);

  dim3 gss(10, 32);    // 256 strips (16x64 each), 8 waves/block
  gram_kernel<<<gss, 256, 0, stream>>>(ws + SUPN_OFF, ws + SNS_OFF,
                                       ws + QRYN_OFF, ws + QN_OFF,
                                       ws + ACCSS_OFF, 0);
  dim3 gqq(150, 7);    // 49 single tiles
  gram_kernel<<<gqq, 256, 0, stream>>>(ws + SUPN_OFF, ws + SNS_OFF,
                                       ws + QRYN_OFF, ws + QN_OFF,
                                       ws + ACCQQ_OFF, 1);
  dim3 gqs(750, 7);    // 56 strips (16x64 each)
  gram_kernel<<<gqs, 256, 0, stream>>>(ws + SUPN_OFF, ws + SNS_OFF,
                                       ws + QRYN_OFF, ws + QN_OFF,
                                       ws + ACCQS_OFF, 2);

  finalize_kernel<<<1, 256, 0, stream>>>(ws, qy, out);
}